// sMPS_7748121002650
// MI455X (gfx1250) — compile-verified
//
#include <hip/hip_runtime.h>
#include <hip/hip_bf16.h>

typedef __attribute__((ext_vector_type(2))) float v2f;
typedef __attribute__((ext_vector_type(8))) float v8f;

#define LDP 65  // padded LDS row stride (floats) -> conflict-free column reads

// One workgroup (4 wave32s) per sample n.
// E evolves as: E <- sum_mu T_mu^T * E * T_mu  (64x64 f32, via v_wmma_f32_16x16x4_f32)
__global__ __launch_bounds__(128) void smps_kernel(
    const float* __restrict__ x,         // (N, 64, 2)
    const float* __restrict__ mps0,      // (4, 1, 2, 64)
    const float* __restrict__ mps_mid,   // (62, 4, 64, 2, 64)
    const float* __restrict__ mps_last,  // (4, 64, 2, 1)
    float* __restrict__ out)             // (N)
{
    __shared__ float E [64 * LDP];
    __shared__ float T [64 * LDP];
    __shared__ float M1[64 * LDP];
    __shared__ float u [4 * 64];
    __shared__ float red[128];

    const int n    = blockIdx.x;
    const int tid  = threadIdx.x;
    const int lane = tid & 31;
    const int wave = tid >> 5;
    const int m    = lane & 15;            // M (A) / N (B,C,D) index within tile
    const int kq   = (lane >> 4) << 1;     // K sub-offset: lanes 0-15 -> K0,K1; 16-31 -> K2,K3
    const int rb   = (lane >> 4) << 3;     // C/D row base: 0 or 8
    const int j    = wave * 16;            // this wave's output column band

    const float* xn = x + (size_t)n * 128;

    // ---------------- site 0: E[c,e] = sum_mu u_mu[c] u_mu[e] ----------------
    {
        const float x0 = xn[0], x1 = xn[1];
        for (int q = tid; q < 256; q += 128) {
            const int mu = q >> 6, d = q & 63;
            u[q] = x0 * mps0[mu * 128 + d] + x1 * mps0[mu * 128 + 64 + d];
        }
        __syncthreads();
        for (int q = tid; q < 4096; q += 128) {
            const int r = q >> 6, c = q & 63;
            float s = 0.f;
            #pragma unroll
            for (int mu = 0; mu < 4; mu++)
                s += u[mu * 64 + r] * u[mu * 64 + c];
            E[r * LDP + c] = s;
        }
    }

    // ---------------- 62 middle sites ----------------
    for (int s = 0; s < 62; s++) {
        const float xi0 = xn[2 * (s + 1)];
        const float xi1 = xn[2 * (s + 1) + 1];
        const float* As = mps_mid + (size_t)s * 4 * 8192;

        v8f cacc[4] = {v8f{}, v8f{}, v8f{}, v8f{}};

        for (int mu = 0; mu < 4; mu++) {
            __syncthreads();  // T/M1 free from previous mu; E ready (from prev site)

            // T = xi0 * A[s,mu,:,0,:] + xi1 * A[s,mu,:,1,:]   (cooperative, 128 threads)
            const float* Amu = As + mu * 8192;
            for (int q = tid; q < 4096; q += 128) {
                const int bb = q >> 6, dd = q & 63;
                T[bb * LDP + dd] = xi0 * Amu[bb * 128 + dd] + xi1 * Amu[bb * 128 + 64 + dd];
            }
            __syncthreads();

            // GEMM1: M1[:, j..j+15] = E * T  (this wave's column band, 4 row tiles)
            #pragma unroll
            for (int ti = 0; ti < 4; ti++) {
                const int i = ti * 16;
                v8f acc = {};
                #pragma unroll
                for (int k0 = 0; k0 < 64; k0 += 4) {
                    v2f a, b;
                    a.x = E[(i + m) * LDP + k0 + kq];        // A[m][kq]   = E[i+m][k]
                    a.y = E[(i + m) * LDP + k0 + kq + 1];    // A[m][kq+1]
                    b.x = T[(k0 + kq) * LDP + j + m];        // B[kq][n]   = T[k][j+n]
                    b.y = T[(k0 + kq + 1) * LDP + j + m];    // B[kq+1][n]
                    acc = __builtin_amdgcn_wmma_f32_16x16x4_f32(
                        false, a, false, b, (short)0, acc, false, false);
                }
                #pragma unroll
                for (int v = 0; v < 8; v++)
                    M1[(i + rb + v) * LDP + j + m] = acc[v];
            }
            __syncthreads();

            // GEMM2: cacc[ti] += T^T(rows f..f+15) * M1[:, j..j+15]
            #pragma unroll
            for (int ti = 0; ti < 4; ti++) {
                const int f = ti * 16;
                v8f acc = cacc[ti];
                #pragma unroll
                for (int k0 = 0; k0 < 64; k0 += 4) {
                    v2f a, b;
                    a.x = T[(k0 + kq) * LDP + f + m];        // A[m][kq]   = T[k][f+m] (transposed read)
                    a.y = T[(k0 + kq + 1) * LDP + f + m];
                    b.x = M1[(k0 + kq) * LDP + j + m];       // B[kq][n]   = M1[k][j+n]
                    b.y = M1[(k0 + kq + 1) * LDP + j + m];
                    acc = __builtin_amdgcn_wmma_f32_16x16x4_f32(
                        false, a, false, b, (short)0, acc, false, false);
                }
                cacc[ti] = acc;
            }
        }

        __syncthreads();  // everyone done with E (GEMM1 of mu=3) and GEMM2
        // write accumulators as new E (each wave owns column band j, exclusive)
        #pragma unroll
        for (int ti = 0; ti < 4; ti++) {
            #pragma unroll
            for (int v = 0; v < 8; v++)
                E[(ti * 16 + rb + v) * LDP + j + m] = cacc[ti][v];
        }
        // top-of-loop __syncthreads() protects E before next site's GEMM1
    }

    // ---------------- last site: p = | sum_mu v_mu^T E v_mu | ----------------
    __syncthreads();
    {
        const float xl0 = xn[126], xl1 = xn[127];
        for (int q = tid; q < 256; q += 128) {
            const int mu = q >> 6, b = q & 63;
            u[q] = xl0 * mps_last[mu * 128 + b * 2] + xl1 * mps_last[mu * 128 + b * 2 + 1];
        }
        __syncthreads();
        float p = 0.f;
        for (int q = tid; q < 16384; q += 128) {
            const int mu = q >> 12, r = (q >> 6) & 63, d = q & 63;
            p += u[mu * 64 + r] * E[r * LDP + d] * u[mu * 64 + d];
        }
        red[tid] = p;
        __syncthreads();
        for (int off = 64; off > 0; off >>= 1) {
            if (tid < off) red[tid] += red[tid + off];
            __syncthreads();
        }
        if (tid == 0) out[n] = fabsf(red[0]);
    }
}

extern "C" void kernel_launch(void* const* d_in, const int* in_sizes, int n_in,
                              void* d_out, int out_size, void* d_ws, size_t ws_size,
                              hipStream_t stream) {
    const float* x        = (const float*)d_in[0];
    const float* mps0     = (const float*)d_in[1];
    const float* mps_mid  = (const float*)d_in[2];
    const float* mps_last = (const float*)d_in[3];
    float* out = (float*)d_out;

    const int N = out_size;  // 2048 samples, one workgroup (4 wave32s) each
    smps_kernel<<<dim3(N), dim3(128), 0, stream>>>(x, mps0, mps_mid, mps_last, out);
}